// Transformer_80779744903184
// MI455X (gfx1250) — compile-verified
//
#include <hip/hip_runtime.h>
#include <math.h>

// ---- problem constants ---------------------------------------------------
#define T_SEQ 2048
#define D_EMB 1024
#define NH    16
#define DHEAD 64
#define NL    4

typedef __attribute__((ext_vector_type(2))) float v2f;
typedef __attribute__((ext_vector_type(8))) float v8f;

// V_WMMA_F32_16X16X4_F32: A 16x4, B 4x16, C/D 16x16, all f32, wave32.
// Fragment layout (ISA 7.12.2): lane = 16*khalf + m (A) / 16*khalf + n (B);
// a.x = A[m, 2*khalf], a.y = A[m, 2*khalf+1]; b.x = B[2*khalf, n], b.y = B[2*khalf+1, n].
// C/D: vgpr v, lane L -> row = v + 8*(L>>4), col = L&15.
__device__ __forceinline__ v8f wmma_f32(v2f a, v2f b, v8f c) {
  return __builtin_amdgcn_wmma_f32_16x16x4_f32(false, a, false, b, (short)0, c,
                                               false, false);
}

__device__ __forceinline__ float gelu_tanh(float x) {
  const float k = 0.7978845608028654f; // sqrt(2/pi)
  float u = k * (x + 0.044715f * x * x * x);
  return 0.5f * x * (1.0f + tanhf(u));
}

// ---- token embedding -----------------------------------------------------
// x[t,d] = obs[t,:]@Wobs[:,d] + bobs[d] + EA[action[t],d]
//        + reward[t]*Wrew[d] + brew[d] + ET[time[t],d]
__global__ __launch_bounds__(256) void embed_k(
    const float* __restrict__ obs, const int* __restrict__ action,
    const float* __restrict__ reward, const int* __restrict__ timei,
    const float* __restrict__ wobs, const float* __restrict__ bobs,
    const float* __restrict__ ea, const float* __restrict__ wrew,
    const float* __restrict__ brew, const float* __restrict__ et,
    float* __restrict__ x) {
  int t = blockIdx.y;
  int d = blockIdx.x * 256 + threadIdx.x;
  float acc = bobs[d] + brew[d] + reward[t] * wrew[d] +
              ea[(size_t)action[t] * D_EMB + d] +
              et[(size_t)timei[t] * D_EMB + d];
  const float* orow = obs + (size_t)t * 64;
#pragma unroll 8
  for (int k = 0; k < 64; ++k) acc += orow[k] * wobs[k * D_EMB + d];
  x[(size_t)t * D_EMB + d] = acc;
}

// ---- layernorm: one wave32 per row of 1024 -------------------------------
__global__ __launch_bounds__(256) void ln_k(const float* __restrict__ x,
                                            const float* __restrict__ scale,
                                            const float* __restrict__ bias,
                                            float* __restrict__ y) {
  int row = blockIdx.x * 8 + (threadIdx.x >> 5);
  int lane = threadIdx.x & 31;
  const float* xr = x + (size_t)row * D_EMB;
  float vals[32];
  float s = 0.f;
#pragma unroll
  for (int i = 0; i < 32; ++i) {
    vals[i] = xr[lane + i * 32];
    s += vals[i];
  }
#pragma unroll
  for (int off = 16; off > 0; off >>= 1) s += __shfl_xor(s, off, 32);
  float mean = s * (1.0f / D_EMB);
  float vs = 0.f;
#pragma unroll
  for (int i = 0; i < 32; ++i) {
    float d = vals[i] - mean;
    vs += d * d;
  }
#pragma unroll
  for (int off = 16; off > 0; off >>= 1) vs += __shfl_xor(vs, off, 32);
  float inv = rsqrtf(vs * (1.0f / D_EMB) + 1e-6f);
  float* yr = y + (size_t)row * D_EMB;
#pragma unroll
  for (int i = 0; i < 32; ++i) {
    int c = lane + i * 32;
    float g = scale ? scale[c] : 1.0f;
    float b = bias ? bias[c] : 0.0f;
    yr[c] = (vals[i] - mean) * inv * g + b;
  }
}

// ---- WMMA f32 GEMM: out[M,N] = A[M,K] @ W[K,N] + bias (+gelu) (+residual)
// block = 256 threads = 8 waves; wave owns 32 rows x 64 cols (2x4 16x16 tiles)
// B panel (64k x 64n) staged in LDS per block, swizzled so each WMMA B
// fragment pair is one ds_load_b64. Next panel prefetched (global_prefetch).
__global__ __launch_bounds__(256) void gemm_wmma_k(
    const float* __restrict__ A, const float* __restrict__ W,
    const float* __restrict__ bias, const float* residual, float* out, int M,
    int N, int K, int act) {
  __shared__ float Bs[64 * 64];  // B[kk][n] at Bs[(kk>>1)*128 + 2n + (kk&1)]
  int tid = threadIdx.x;
  int lane = tid & 31;
  int wave = tid >> 5;
  int r = lane & 15;     // m-row of A fragment / n-col of B fragment
  int half = lane >> 4;  // khalf
  int m0 = blockIdx.y * 256 + wave * 32;
  int n0 = blockIdx.x * 64;
  v8f acc[2][4] = {};
  const float* a0p = A + (size_t)(m0 + r) * K;
  const float* a1p = A + (size_t)(m0 + 16 + r) * K;
  for (int kb = 0; kb < K; kb += 64) {
    __syncthreads();  // previous tile's ds reads done before overwrite
#pragma unroll
    for (int i = 0; i < 16; ++i) {
      int idx = tid + i * 256;
      int kk = idx >> 6, n = idx & 63;  // coalesced in n
      const float* gp = &W[(size_t)(kb + kk) * N + n0 + n];
      Bs[(kk >> 1) * 128 + (n << 1) + (kk & 1)] = *gp;
      if (kb + 64 < K && (n & 15) == 0)  // one prefetch per 64B line
        __builtin_prefetch(gp + (size_t)64 * N, 0, 0);
    }
    __syncthreads();
#pragma unroll
    for (int k0 = 0; k0 < 64; k0 += 4) {
      v2f a0 = *(const v2f*)(a0p + kb + k0 + 2 * half);
      v2f a1 = *(const v2f*)(a1p + kb + k0 + 2 * half);
      int krow = (k0 + 2 * half) >> 1;  // kk even -> pair row in Bs
#pragma unroll
      for (int j = 0; j < 4; ++j) {
        v2f b = *(const v2f*)(&Bs[krow * 128 + ((j * 16 + r) << 1)]);
        acc[0][j] = wmma_f32(a0, b, acc[0][j]);
        acc[1][j] = wmma_f32(a1, b, acc[1][j]);
      }
    }
  }
#pragma unroll
  for (int t = 0; t < 2; ++t)
#pragma unroll
    for (int j = 0; j < 4; ++j)
#pragma unroll
      for (int v = 0; v < 8; ++v) {
        int row = m0 + t * 16 + v + 8 * half;
        int col = n0 + j * 16 + r;
        float val = acc[t][j][v] + bias[col];
        if (act) val = gelu_tanh(val);
        if (residual) val += residual[(size_t)row * N + col];
        out[(size_t)row * N + col] = val;
      }
}

// ---- causal masked attention (no softmax), one wave per (head, q-tile) ---
// kqv: [T, 3*D] with flax split order K | Q | V.  out: [T, D] (head-major cols)
__global__ __launch_bounds__(32) void attn_wmma_k(const float* __restrict__ kqv,
                                                  float* __restrict__ out) {
  __shared__ float Sld[16 * 16];
  int qt = blockIdx.x;
  int h = blockIdx.y;
  int lane = threadIdx.x;
  int r = lane & 15;
  int half = lane >> 4;
  const int ld = 3 * D_EMB;
  const float* Kp = kqv + h * DHEAD;              // K block
  const float* Qp = kqv + D_EMB + h * DHEAD;      // Q block
  const float* Vp = kqv + 2 * D_EMB + h * DHEAD;  // V block
  v8f o[4] = {};
  for (int st = 0; st <= qt; ++st) {
    // S = Q_tile(16x64) @ K_tile^T(64x16); B[k,n] = K[st*16+n, k]
    v8f s = {};
#pragma unroll
    for (int kc = 0; kc < DHEAD; kc += 4) {
      v2f a = *(const v2f*)(Qp + (size_t)(qt * 16 + r) * ld + kc + 2 * half);
      v2f b = *(const v2f*)(Kp + (size_t)(st * 16 + r) * ld + kc + 2 * half);
      s = wmma_f32(a, b, s);
    }
    if (st == qt) {  // causal mask on the diagonal tile: keep col<=row
#pragma unroll
      for (int v = 0; v < 8; ++v) {
        int m = v + 8 * half;
        if (r > m) s[v] = 0.0f;
      }
    }
    // C-layout -> A-layout via LDS round trip
    __syncthreads();
#pragma unroll
    for (int v = 0; v < 8; ++v) Sld[(v + 8 * half) * 16 + r] = s[v];
    __syncthreads();
    // O(16x64) += S(16x16) @ V_tile(16x64)
#pragma unroll
    for (int jc = 0; jc < 4; ++jc) {
      v2f sa = *(const v2f*)(&Sld[r * 16 + jc * 4 + 2 * half]);
#pragma unroll
      for (int j = 0; j < 4; ++j) {
        const float* vp =
            Vp + (size_t)(st * 16 + jc * 4 + 2 * half) * ld + j * 16 + r;
        v2f b;
        b.x = vp[0];
        b.y = vp[ld];
        o[j] = wmma_f32(sa, b, o[j]);
      }
    }
  }
#pragma unroll
  for (int j = 0; j < 4; ++j)
#pragma unroll
    for (int v = 0; v < 8; ++v) {
      int t = qt * 16 + v + 8 * half;
      int c = h * DHEAD + j * 16 + r;
      out[(size_t)t * D_EMB + c] = o[j][v];
    }
}

// ---- actor / critic heads (N=10 and N=1: below WMMA granularity) ---------
__global__ __launch_bounds__(256) void heads_k(
    const float* __restrict__ A, const float* __restrict__ wa,
    const float* __restrict__ ba, const float* __restrict__ wc,
    const float* __restrict__ bc, float* __restrict__ out) {
  int idx = blockIdx.x * 256 + threadIdx.x;
  if (idx >= T_SEQ * 11) return;
  int t = idx / 11, j = idx % 11;
  const float* ar = A + (size_t)t * D_EMB;
  if (j < 10) {
    float s = ba[j];
    for (int k = 0; k < D_EMB; ++k) s += ar[k] * wa[k * 10 + j];
    out[t * 10 + j] = s;
  } else {
    float s = bc[0];
    for (int k = 0; k < D_EMB; ++k) s += ar[k] * wc[k];
    out[T_SEQ * 10 + t] = s;
  }
}

// ---- host-side orchestration ---------------------------------------------
extern "C" void kernel_launch(void* const* d_in, const int* in_sizes, int n_in,
                              void* d_out, int out_size, void* d_ws,
                              size_t ws_size, hipStream_t stream) {
  (void)in_sizes; (void)n_in; (void)out_size; (void)ws_size;
  // jax tree_leaves order (dict keys sorted at every level):
  // 0 obs, 1 action, 2 reward, 3 time,
  // params: 4 actor.b, 5 actor.w,
  //         blocks[l] @ 6+12*l: ln1.bias, ln1.scale, ln2.bias, ln2.scale,
  //                             mha.d1.b, mha.d1.w, mha.d2.b, mha.d2.w,
  //                             mlp.d1.b, mlp.d1.w, mlp.d2.b, mlp.d2.w,
  //         54 critic.b, 55 critic.w, 56 embed_action,
  //         57 embed_obs.b, 58 embed_obs.w, 59 embed_reward.b,
  //         60 embed_reward.w, 61 embed_time
  const float* obs = (const float*)d_in[0];
  const int* action = (const int*)d_in[1];
  const float* reward = (const float*)d_in[2];
  const int* timei = (const int*)d_in[3];
  const float* actor_b = (const float*)d_in[4];
  const float* actor_w = (const float*)d_in[5];
  struct Blk {
    const float *ln1b, *ln1s, *ln2b, *ln2s;
    const float *d1b, *d1w, *d2b, *d2w;
    const float *m1b, *m1w, *m2b, *m2w;
  } blk[NL];
  for (int l = 0; l < NL; ++l) {
    int b = 6 + l * 12;
    blk[l].ln1b = (const float*)d_in[b + 0];
    blk[l].ln1s = (const float*)d_in[b + 1];
    blk[l].ln2b = (const float*)d_in[b + 2];
    blk[l].ln2s = (const float*)d_in[b + 3];
    blk[l].d1b = (const float*)d_in[b + 4];
    blk[l].d1w = (const float*)d_in[b + 5];
    blk[l].d2b = (const float*)d_in[b + 6];
    blk[l].d2w = (const float*)d_in[b + 7];
    blk[l].m1b = (const float*)d_in[b + 8];
    blk[l].m1w = (const float*)d_in[b + 9];
    blk[l].m2b = (const float*)d_in[b + 10];
    blk[l].m2w = (const float*)d_in[b + 11];
  }
  const float* critic_b = (const float*)d_in[54];
  const float* critic_w = (const float*)d_in[55];
  const float* emb_act = (const float*)d_in[56];
  const float* obs_b = (const float*)d_in[57];
  const float* obs_w = (const float*)d_in[58];
  const float* rew_b = (const float*)d_in[59];
  const float* rew_w = (const float*)d_in[60];
  const float* emb_time = (const float*)d_in[61];

  float* ws = (float*)d_ws;
  float* X = ws;                                  // [T, D] residual stream
  float* Abuf = X + (size_t)T_SEQ * D_EMB;        // [T, D] LN output
  float* Bbuf = Abuf + (size_t)T_SEQ * D_EMB;     // [T, 4D] kqv / mlp hidden
  float* Cbuf = Bbuf + (size_t)T_SEQ * 4 * D_EMB; // [T, D] attention out

  embed_k<<<dim3(D_EMB / 256, T_SEQ), 256, 0, stream>>>(
      obs, action, reward, timei, obs_w, obs_b, emb_act, rew_w, rew_b, emb_time,
      X);

  for (int l = 0; l < NL; ++l) {
    ln_k<<<T_SEQ / 8, 256, 0, stream>>>(X, blk[l].ln1s, blk[l].ln1b, Abuf);
    gemm_wmma_k<<<dim3((3 * D_EMB) / 64, T_SEQ / 256), 256, 0, stream>>>(
        Abuf, blk[l].d1w, blk[l].d1b, nullptr, Bbuf, T_SEQ, 3 * D_EMB, D_EMB,
        0);
    attn_wmma_k<<<dim3(T_SEQ / 16, NH), 32, 0, stream>>>(Bbuf, Cbuf);
    gemm_wmma_k<<<dim3(D_EMB / 64, T_SEQ / 256), 256, 0, stream>>>(
        Cbuf, blk[l].d2w, blk[l].d2b, X, X, T_SEQ, D_EMB, D_EMB, 0);
    ln_k<<<T_SEQ / 8, 256, 0, stream>>>(X, blk[l].ln2s, blk[l].ln2b, Abuf);
    gemm_wmma_k<<<dim3((4 * D_EMB) / 64, T_SEQ / 256), 256, 0, stream>>>(
        Abuf, blk[l].m1w, blk[l].m1b, nullptr, Bbuf, T_SEQ, 4 * D_EMB, D_EMB,
        1);
    gemm_wmma_k<<<dim3(D_EMB / 64, T_SEQ / 256), 256, 0, stream>>>(
        Bbuf, blk[l].m2w, blk[l].m2b, X, X, T_SEQ, D_EMB, 4 * D_EMB, 0);
  }
  ln_k<<<T_SEQ / 8, 256, 0, stream>>>(X, nullptr, nullptr, Abuf);
  heads_k<<<(T_SEQ * 11 + 255) / 256, 256, 0, stream>>>(
      Abuf, actor_w, actor_b, critic_w, critic_b, (float*)d_out);
}